// HMLDM_37065567765084
// MI455X (gfx1250) — compile-verified
//
#include <hip/hip_runtime.h>

#define EPSF   1e-6f
#define LOG2E  1.4426950408889634f

typedef __attribute__((ext_vector_type(2))) float v2f;
typedef __attribute__((ext_vector_type(8))) float v8f;

// ---------------------------------------------------------------------------
// Kernel 1: per-row softmax (D=8), -2*w copy, squared norms, gamma*log2e
// ---------------------------------------------------------------------------
__global__ __launch_bounds__(256) void softmax_kernel(
    const float* __restrict__ gamma, const float* __restrict__ lw,
    float* __restrict__ w, float* __restrict__ wm2, float* __restrict__ sq,
    float* __restrict__ gl, int N) {
  int i = blockIdx.x * blockDim.x + threadIdx.x;
  if (i >= N) return;
  const float4* p = (const float4*)(lw + (size_t)i * 8);
  float4 a = p[0], b = p[1];
  float x[8] = {a.x, a.y, a.z, a.w, b.x, b.y, b.z, b.w};
  float m = x[0];
#pragma unroll
  for (int k = 1; k < 8; ++k) m = fmaxf(m, x[k]);
  float s = 0.f;
#pragma unroll
  for (int k = 0; k < 8; ++k) { x[k] = __expf(x[k] - m); s += x[k]; }
  float inv = 1.0f / s;
  float q = 0.f;
#pragma unroll
  for (int k = 0; k < 8; ++k) { x[k] *= inv; q += x[k] * x[k]; }
  float4* o = (float4*)(w + (size_t)i * 8);
  o[0] = make_float4(x[0], x[1], x[2], x[3]);
  o[1] = make_float4(x[4], x[5], x[6], x[7]);
  float4* o2 = (float4*)(wm2 + (size_t)i * 8);
  o2[0] = make_float4(-2.f * x[0], -2.f * x[1], -2.f * x[2], -2.f * x[3]);
  o2[1] = make_float4(-2.f * x[4], -2.f * x[5], -2.f * x[6], -2.f * x[7]);
  sq[i] = q;
  gl[i] = gamma[i] * LOG2E;
}

// ---------------------------------------------------------------------------
// Kernel 2: upper-triangular pairwise term.
// 3x V_WMMA_F32_16X16X4_F32 per 16x16 tile: accumulator = sq_i + sq_j - 2*dot.
// Epilogue: exp2(L*(g_i + g_j - eps) - L*dist) summed over i < j.
// Grid: (N/16, N/128), 256 threads = 8 waves, one tile per wave.
// ---------------------------------------------------------------------------
__global__ __launch_bounds__(256) void pair_kernel(
    const float* __restrict__ w, const float* __restrict__ wm2,
    const float* __restrict__ sq, const float* __restrict__ gl,
    float* __restrict__ partials, int N) {
  const int lane = threadIdx.x & 31;
  const int wave = threadIdx.x >> 5;
  const int l16  = lane & 15;
  const int half = lane >> 4;

  const int i0  = blockIdx.x * 16;    // row-tile base
  const int jb0 = blockIdx.y * 128;   // block col-strip base
  const size_t pidx = (size_t)blockIdx.y * gridDim.x + blockIdx.x;

  // whole block strictly below the diagonal -> zero partial
  if (jb0 + 128 <= i0) {
    if (threadIdx.x == 0) partials[pidx] = 0.f;
    return;
  }

  const int j0 = jb0 + wave * 16;     // this wave's col-tile base
  float acc0 = 0.f, acc1 = 0.f;

  if (j0 + 16 > i0) {   // wave's tile touches the upper triangle
    // A fragments from -2*w (rows i0..i0+15), B fragments from w (rows j0..)
    const float* wa = wm2 + (size_t)(i0 + l16) * 8 + half * 2;
    const float* wb = w   + (size_t)(j0 + l16) * 8 + half * 2;
    v2f a0 = {wa[0], wa[1]};
    v2f b0 = {wb[0], wb[1]};
    v2f a1 = {wa[4], wa[5]};
    v2f b1 = {wb[4], wb[5]};

    // rank-2 correction fragments: C += sq_i * 1 + 1 * sq_j
    const float sqi = sq[i0 + l16];
    const float sqj = sq[j0 + l16];
    v2f a2, b2;
    a2.x = half ? 0.f : sqi;  a2.y = half ? 0.f : 1.0f;
    b2.x = half ? 0.f : 1.0f; b2.y = half ? 0.f : sqj;

    v8f c = {};
    c = __builtin_amdgcn_wmma_f32_16x16x4_f32(false, a0, false, b0, (short)0,
                                              c, false, false);
    c = __builtin_amdgcn_wmma_f32_16x16x4_f32(false, a1, false, b1, (short)0,
                                              c, false, false);
    c = __builtin_amdgcn_wmma_f32_16x16x4_f32(false, a2, false, b2, (short)0,
                                              c, false, false);

    const int   jcol   = j0 + l16;
    const float base_j = gl[jcol] - (LOG2E * EPSF);  // L*(g_j - eps)
    const float gli    = gl[i0 + l16];               // lane-held row data

    if (i0 == j0) {
      // diagonal tile: keep strictly upper-triangular elements
#pragma unroll
      for (int r = 0; r < 8; ++r) {
        const float glm  = __shfl(gli, r + half * 8, 32);
        const float d2   = fmaxf(c[r], 0.f);
        const float dist = __builtin_amdgcn_sqrtf(d2);
        const float val  = __builtin_amdgcn_exp2f(
            fmaf(dist, -LOG2E, glm + base_j));
        const int m = i0 + r + half * 8;
        if (m < jcol) { if (r & 1) acc1 += val; else acc0 += val; }
      }
    } else {
      // fully above the diagonal: unmasked
#pragma unroll
      for (int r = 0; r < 8; ++r) {
        const float glm  = __shfl(gli, r + half * 8, 32);
        const float d2   = fmaxf(c[r], 0.f);
        const float dist = __builtin_amdgcn_sqrtf(d2);
        const float val  = __builtin_amdgcn_exp2f(
            fmaf(dist, -LOG2E, glm + base_j));
        if (r & 1) acc1 += val; else acc0 += val;
      }
    }
  }

  float acc = acc0 + acc1;
#pragma unroll
  for (int off = 16; off > 0; off >>= 1) acc += __shfl_down(acc, off, 32);

  __shared__ float red[8];
  if (lane == 0) red[wave] = acc;
  __syncthreads();
  if (threadIdx.x == 0) {
    float t = 0.f;
#pragma unroll
    for (int k = 0; k < 8; ++k) t += red[k];
    partials[pidx] = t;
  }
}

// ---------------------------------------------------------------------------
// Kernel 3: edge (link) term, grid-stride, per-block partials
// ---------------------------------------------------------------------------
__global__ __launch_bounds__(256) void edge_kernel(
    const float* __restrict__ w, const float* __restrict__ gamma,
    const int* __restrict__ ei, const int* __restrict__ ej,
    float* __restrict__ partials, int E) {
  float acc = 0.f;
  for (int e = blockIdx.x * blockDim.x + threadIdx.x; e < E;
       e += gridDim.x * blockDim.x) {
    int i = ei[e], j = ej[e];
    const float4* pi = (const float4*)(w + (size_t)i * 8);
    const float4* pj = (const float4*)(w + (size_t)j * 8);
    float4 a0 = pi[0], a1 = pi[1], b0 = pj[0], b1 = pj[1];
    float d, s = 0.f;
    d = a0.x - b0.x + EPSF; s += d * d;
    d = a0.y - b0.y + EPSF; s += d * d;
    d = a0.z - b0.z + EPSF; s += d * d;
    d = a0.w - b0.w + EPSF; s += d * d;
    d = a1.x - b1.x + EPSF; s += d * d;
    d = a1.y - b1.y + EPSF; s += d * d;
    d = a1.z - b1.z + EPSF; s += d * d;
    d = a1.w - b1.w + EPSF; s += d * d;
    acc += -__builtin_amdgcn_sqrtf(s) + gamma[i] + gamma[j];  // DELTA = 1
  }
#pragma unroll
  for (int off = 16; off > 0; off >>= 1) acc += __shfl_down(acc, off, 32);
  __shared__ float red[8];
  int lane = threadIdx.x & 31, wv = threadIdx.x >> 5;
  if (lane == 0) red[wv] = acc;
  __syncthreads();
  if (threadIdx.x == 0) {
    float t = 0.f;
#pragma unroll
    for (int k = 0; k < 8; ++k) t += red[k];
    partials[blockIdx.x] = t;
  }
}

// ---------------------------------------------------------------------------
// Kernel 4: final reduction. Upper-triangle sum already equals 0.5*egᵀMat eg,
// so out = S_pair - S_edge.
// ---------------------------------------------------------------------------
__global__ __launch_bounds__(256) void final_kernel(
    const float* __restrict__ pairP, int np, const float* __restrict__ edgeP,
    int ne, float* __restrict__ out) {
  float a1 = 0.f, a2 = 0.f;
  for (int k = threadIdx.x; k < np; k += 256) a1 += pairP[k];
  for (int k = threadIdx.x; k < ne; k += 256) a2 += edgeP[k];
#pragma unroll
  for (int off = 16; off > 0; off >>= 1) {
    a1 += __shfl_down(a1, off, 32);
    a2 += __shfl_down(a2, off, 32);
  }
  __shared__ float r1[8], r2[8];
  int lane = threadIdx.x & 31, wv = threadIdx.x >> 5;
  if (lane == 0) { r1[wv] = a1; r2[wv] = a2; }
  __syncthreads();
  if (threadIdx.x == 0) {
    float s1 = 0.f, s2 = 0.f;
#pragma unroll
    for (int k = 0; k < 8; ++k) { s1 += r1[k]; s2 += r2[k]; }
    out[0] = s1 - s2;
  }
}

// ---------------------------------------------------------------------------
extern "C" void kernel_launch(void* const* d_in, const int* in_sizes, int n_in,
                              void* d_out, int out_size, void* d_ws,
                              size_t ws_size, hipStream_t stream) {
  const float* gamma = (const float*)d_in[0];
  const float* lw    = (const float*)d_in[1];
  const int*   ei    = (const int*)d_in[2];
  const int*   ej    = (const int*)d_in[3];
  float* out = (float*)d_out;

  const int N = in_sizes[0];      // 16384
  const int E = in_sizes[2];      // 524288

  // workspace layout (floats)
  float* w     = (float*)d_ws;            // N*8
  float* wm2   = w + (size_t)N * 8;       // N*8  (-2*w)
  float* sq    = wm2 + (size_t)N * 8;     // N
  float* gl    = sq + N;                  // N    (gamma * log2e)
  const int gx = N / 16;                  // row tiles
  const int gy = N / 128;                 // col strips (8 waves * 16)
  const int np = gx * gy;
  float* pairP = gl + N;                  // np
  const int ne = 1024;
  float* edgeP = pairP + np;              // ne

  softmax_kernel<<<(N + 255) / 256, 256, 0, stream>>>(gamma, lw, w, wm2, sq,
                                                      gl, N);
  pair_kernel<<<dim3(gx, gy), 256, 0, stream>>>(w, wm2, sq, gl, pairP, N);
  edge_kernel<<<ne, 256, 0, stream>>>(w, gamma, ei, ej, edgeP, E);
  final_kernel<<<1, 256, 0, stream>>>(pairP, np, edgeP, ne, out);
}